// Dp_Agg_1898375545031
// MI455X (gfx1250) — compile-verified
//
#include <hip/hip_runtime.h>
#include <math.h>

typedef float v4f __attribute__((ext_vector_type(4)));

#define EMB_DIM 64
#define LANES_PER_ROW 16        // 64 floats / 4 per lane; half a wave32 per row
#define BLK 256                 // 8 wave32 per block

#define SCAN_BLK 256
#define SCAN_EPT 4
#define SCAN_TILE (SCAN_BLK * SCAN_EPT)   // 1024 elements per scan block

// ===========================================================================
// Shared pass: histogram  counts[loc] += 1  (u32 atomic, no return)
// ===========================================================================
__global__ void dp_zero_counts(unsigned int* __restrict__ counts, int n_locs) {
    int i = blockIdx.x * blockDim.x + threadIdx.x;
    if (i < n_locs) counts[i] = 0u;
}

__global__ void dp_count_edges(const int* __restrict__ edge_locs,
                               unsigned int* __restrict__ counts,
                               int n_edges) {
    int e = blockIdx.x * blockDim.x + threadIdx.x;
    if (e >= n_edges) return;
    atomicAdd(&counts[edge_locs[e]], 1u);
}

// ===========================================================================
// CSR path, step 1: block-level exclusive scan of counts (1024 elems / block).
// LDS (ds_*) Hillis-Steele scan across the 256 thread partial sums.
// ===========================================================================
__global__ void dp_scan_blocks(const unsigned int* __restrict__ counts,
                               unsigned int* __restrict__ starts,
                               unsigned int* __restrict__ bsums,
                               int n) {
    __shared__ unsigned int sh[SCAN_BLK];
    const int t = threadIdx.x;
    const int base = blockIdx.x * SCAN_TILE + t * SCAN_EPT;

    unsigned int v0 = (base + 0 < n) ? counts[base + 0] : 0u;
    unsigned int v1 = (base + 1 < n) ? counts[base + 1] : 0u;
    unsigned int v2 = (base + 2 < n) ? counts[base + 2] : 0u;
    unsigned int v3 = (base + 3 < n) ? counts[base + 3] : 0u;
    const unsigned int tsum = v0 + v1 + v2 + v3;

    sh[t] = tsum;
    __syncthreads();
    for (int off = 1; off < SCAN_BLK; off <<= 1) {
        unsigned int u = (t >= off) ? sh[t - off] : 0u;
        __syncthreads();
        sh[t] += u;
        __syncthreads();
    }
    unsigned int p = sh[t] - tsum;                    // exclusive prefix of this thread
    if (t == SCAN_BLK - 1) bsums[blockIdx.x] = sh[t]; // block total

    if (base + 0 < n) { starts[base + 0] = p; p += v0; }
    if (base + 1 < n) { starts[base + 1] = p; p += v1; }
    if (base + 2 < n) { starts[base + 2] = p; p += v2; }
    if (base + 3 < n) { starts[base + 3] = p; }
}

// Step 2: single-block in-place exclusive scan of the block sums (n <= 1024).
__global__ void dp_scan_bsums(unsigned int* __restrict__ bsums, int n) {
    __shared__ unsigned int sh[SCAN_BLK];
    const int t = threadIdx.x;
    const int base = t * SCAN_EPT;

    unsigned int v0 = (base + 0 < n) ? bsums[base + 0] : 0u;
    unsigned int v1 = (base + 1 < n) ? bsums[base + 1] : 0u;
    unsigned int v2 = (base + 2 < n) ? bsums[base + 2] : 0u;
    unsigned int v3 = (base + 3 < n) ? bsums[base + 3] : 0u;
    const unsigned int tsum = v0 + v1 + v2 + v3;

    sh[t] = tsum;
    __syncthreads();
    for (int off = 1; off < SCAN_BLK; off <<= 1) {
        unsigned int u = (t >= off) ? sh[t - off] : 0u;
        __syncthreads();
        sh[t] += u;
        __syncthreads();
    }
    unsigned int p = sh[t] - tsum;

    if (base + 0 < n) { bsums[base + 0] = p; p += v0; }
    if (base + 1 < n) { bsums[base + 1] = p; p += v1; }
    if (base + 2 < n) { bsums[base + 2] = p; p += v2; }
    if (base + 3 < n) { bsums[base + 3] = p; }
}

// Step 3: add scanned block sums back; also seed the fill cursors.
__global__ void dp_scan_finalize(unsigned int* __restrict__ starts,
                                 unsigned int* __restrict__ cursors,
                                 const unsigned int* __restrict__ bsums,
                                 int n) {
    int i = blockIdx.x * blockDim.x + threadIdx.x;
    if (i >= n) return;
    unsigned int v = starts[i] + bsums[i / SCAN_TILE];
    starts[i]  = v;
    cursors[i] = v;
}

// Step 4: bucket-fill edge ids (u32 atomic WITH return -> LOADcnt path).
__global__ void dp_fill_edges(const int* __restrict__ edge_locs,
                              unsigned int* __restrict__ cursors,
                              unsigned int* __restrict__ edge_ids,
                              int n_edges) {
    int e = blockIdx.x * blockDim.x + threadIdx.x;
    if (e >= n_edges) return;
    unsigned int pos = atomicAdd(&cursors[edge_locs[e]], 1u);
    edge_ids[pos] = (unsigned int)e;
}

// ===========================================================================
// CSR path, step 5: gather. 16 lanes per location, float4 accumulator per
// lane. Scale sigma/cnt hoisted out of the loop (loc-uniform). Single
// coalesced GLOBAL_LOAD_B128 per row chunk, single GLOBAL_STORE_B128 out,
// zero float atomics. NT loads keep the 1 GB noise stream from thrashing L2.
// ===========================================================================
__global__ void dp_gather(const float* __restrict__ noise,
                          const float* __restrict__ loc_emb,
                          const unsigned int* __restrict__ counts,
                          const unsigned int* __restrict__ starts,
                          const unsigned int* __restrict__ edge_ids,
                          float* __restrict__ out,
                          float sigma, int n_locs) {
    int tid  = blockIdx.x * blockDim.x + threadIdx.x;
    int loc  = tid >> 4;
    int lane = tid & (LANES_PER_ROW - 1);
    if (loc >= n_locs) return;

    const unsigned int cnt   = counts[loc];
    const unsigned int start = starts[loc];
    const unsigned int end   = start + cnt;

    v4f acc = {0.f, 0.f, 0.f, 0.f};
    for (unsigned int j = start; j < end; ++j) {
        unsigned int e = edge_ids[j];
        if (j + 1 < end) {   // hide the random-row latency (global_prefetch_b8)
            __builtin_prefetch(noise + (size_t)edge_ids[j + 1] * EMB_DIM, 0, 0);
        }
        const v4f* nrow = (const v4f*)(noise + (size_t)e * EMB_DIM);
        acc += __builtin_nontemporal_load(&nrow[lane]);
    }

    const float scale = cnt ? (sigma / (float)cnt) : 0.0f;
    const v4f* erow = (const v4f*)(loc_emb + (size_t)loc * EMB_DIM);
    v4f base = __builtin_nontemporal_load(&erow[lane]);
    v4f res = base + acc * scale;
    ((v4f*)(out + (size_t)loc * EMB_DIM))[lane] = res;
}

// ===========================================================================
// Fallback path (small ws): direct atomic scatter (Round-0 design).
// ===========================================================================
__global__ void dp_init_out(const float* __restrict__ loc_emb,
                            float* __restrict__ out, int n_vec4) {
    int i = blockIdx.x * blockDim.x + threadIdx.x;
    if (i >= n_vec4) return;
    v4f v = __builtin_nontemporal_load(&((const v4f*)loc_emb)[i]);
    ((v4f*)out)[i] = v;
}

__global__ void dp_scatter_noise(const float* __restrict__ noise,
                                 const int* __restrict__ edge_locs,
                                 const unsigned int* __restrict__ counts,
                                 float* __restrict__ out,
                                 float sigma, int n_edges) {
    int tid  = blockIdx.x * blockDim.x + threadIdx.x;
    int e    = tid >> 4;
    int lane = tid & (LANES_PER_ROW - 1);
    if (e >= n_edges) return;

    int loc = edge_locs[e];
    float scale = sigma / (float)counts[loc];
    const v4f* nrow = (const v4f*)(noise + (size_t)e * EMB_DIM);
    v4f v = __builtin_nontemporal_load(&nrow[lane]);

    float* orow = out + (size_t)loc * EMB_DIM + (size_t)lane * 4;
    unsafeAtomicAdd(orow + 0, v.x * scale);
    unsafeAtomicAdd(orow + 1, v.y * scale);
    unsafeAtomicAdd(orow + 2, v.z * scale);
    unsafeAtomicAdd(orow + 3, v.w * scale);
}

// ===========================================================================
// Host launcher
// ===========================================================================
extern "C" void kernel_launch(void* const* d_in, const int* in_sizes, int n_in,
                              void* d_out, int out_size, void* d_ws, size_t ws_size,
                              hipStream_t stream) {
    const float* loc_emb   = (const float*)d_in[0];
    const float* noise     = (const float*)d_in[1];
    const int*   edge_locs = (const int*)d_in[2];
    float* out = (float*)d_out;

    const int n_locs  = in_sizes[0] / EMB_DIM;
    const int n_edges = in_sizes[2];

    const double EPS_ = 1.0, DELT_ = 1e-5, CLIP_ = 1.0;
    const float sigma = (float)(CLIP_ * sqrt(2.0 * log(1.25 / DELT_)) / EPS_);

    const int n_scan_blocks = (n_locs + SCAN_TILE - 1) / SCAN_TILE;

    // Carve workspace (256B-aligned regions).
    char* ws = (char*)d_ws;
    size_t off = 0;
    auto carve = [&](size_t bytes) -> char* {
        char* p = ws + off;
        off = (off + bytes + 255) & ~(size_t)255;
        return p;
    };
    unsigned int* counts   = (unsigned int*)carve((size_t)n_locs * 4);
    unsigned int* starts   = (unsigned int*)carve((size_t)n_locs * 4);
    unsigned int* cursors  = (unsigned int*)carve((size_t)n_locs * 4);
    unsigned int* bsums    = (unsigned int*)carve((size_t)n_scan_blocks * 4);
    unsigned int* edge_ids = (unsigned int*)carve((size_t)n_edges * 4);
    const size_t ws_needed = off;

    const bool csr_ok = (ws_size >= ws_needed) && (n_scan_blocks <= SCAN_TILE);

    // Histogram (common to both paths).
    dp_zero_counts<<<(n_locs + BLK - 1) / BLK, BLK, 0, stream>>>(counts, n_locs);
    dp_count_edges<<<(n_edges + BLK - 1) / BLK, BLK, 0, stream>>>(edge_locs, counts, n_edges);

    if (csr_ok) {
        // Exclusive scan of counts -> starts.
        dp_scan_blocks<<<n_scan_blocks, SCAN_BLK, 0, stream>>>(counts, starts, bsums, n_locs);
        dp_scan_bsums<<<1, SCAN_BLK, 0, stream>>>(bsums, n_scan_blocks);
        dp_scan_finalize<<<(n_locs + BLK - 1) / BLK, BLK, 0, stream>>>(starts, cursors, bsums, n_locs);
        // Bucket fill.
        dp_fill_edges<<<(n_edges + BLK - 1) / BLK, BLK, 0, stream>>>(edge_locs, cursors, edge_ids, n_edges);
        // Gather: 16 lanes per location; no float atomics, single store per row.
        int total = n_locs * LANES_PER_ROW;
        dp_gather<<<(total + BLK - 1) / BLK, BLK, 0, stream>>>(
            noise, loc_emb, counts, starts, edge_ids, out, sigma, n_locs);
    } else {
        // Fallback: atomic scatter (only 4 MB of ws needed).
        int n_vec4 = out_size / 4;
        dp_init_out<<<(n_vec4 + BLK - 1) / BLK, BLK, 0, stream>>>(loc_emb, out, n_vec4);
        int total = n_edges * LANES_PER_ROW;
        dp_scatter_noise<<<(total + BLK - 1) / BLK, BLK, 0, stream>>>(
            noise, edge_locs, counts, out, sigma, n_edges);
    }
}